// IoULoss_29351806501126
// MI455X (gfx1250) — compile-verified
//
#include <hip/hip_runtime.h>

typedef float v2f __attribute__((ext_vector_type(2)));
typedef float v8f __attribute__((ext_vector_type(8)));

#define EPSF 1e-7f
// 4 / pi^2
#define FOUR_OVER_PI2 0.405284734569351087f

// Full-wave (32 lane) fp32 sum using V_WMMA_F32_16X16X4_F32.
// A-matrix 16x4 layout (f32): lanes 0-15 -> M=lane, VGPR0=K0, VGPR1=K1;
// lanes 16-31 -> M=lane-16, VGPR0=K2, VGPR1=K3.
// With A[.][0]=p, A[.][1]=0 and B = all-ones, D[m][n] = p_m + p_{m+16}.
// Summing the 8 D VGPRs in a lane gives the half-sum for its lane group;
// one SWAPX16 ds_swizzle completes the 32-lane total (broadcast to all lanes).
// B = ones makes this independent of the exact B 4x16 lane mapping.
__device__ __forceinline__ float wave_sum_wmma(float p) {
    v2f a; a[0] = p;    a[1] = 0.0f;
    v2f b; b[0] = 1.0f; b[1] = 1.0f;
    v8f c = {};
    v8f d = __builtin_amdgcn_wmma_f32_16x16x4_f32(
        /*neg_a=*/false, a, /*neg_b=*/false, b,
        /*c_mod=*/(short)0, c, /*reuse_a=*/false, /*reuse_b=*/false);
    float s = ((d[0] + d[1]) + (d[2] + d[3])) + ((d[4] + d[5]) + (d[6] + d[7]));
    // SWAPX16: group-of-32 swizzle, offset = {xor=0x10, or=0x00, and=0x1f}
    int si = __builtin_amdgcn_ds_swizzle(__float_as_int(s), 0x401F);
    s += __int_as_float(si);
    return s;
}

__device__ __forceinline__ float ciou_loss_elem(float4 P, float4 T) {
    const float x1 = P.x, y1 = P.y, w1 = P.z, h1 = P.w;
    const float x2 = T.x, y2 = T.y, w2 = T.z, h2 = T.w;

    const float w1h = 0.5f * w1, h1h = 0.5f * h1;
    const float w2h = 0.5f * w2, h2h = 0.5f * h2;

    const float b1x1 = x1 - w1h, b1x2 = x1 + w1h;
    const float b1y1 = y1 - h1h, b1y2 = y1 + h1h;
    const float b2x1 = x2 - w2h, b2x2 = x2 + w2h;
    const float b2y1 = y2 - h2h, b2y2 = y2 + h2h;

    const float iw = fmaxf(fminf(b1x2, b2x2) - fmaxf(b1x1, b2x1), 0.0f);
    const float ih = fmaxf(fminf(b1y2, b2y2) - fmaxf(b1y1, b2y1), 0.0f);
    const float inter  = iw * ih;
    const float uni    = w1 * h1 + w2 * h2 - inter + EPSF;
    const float iou    = inter / uni;

    const float cw = fmaxf(b1x2, b2x2) - fminf(b1x1, b2x1);
    const float ch = fmaxf(b1y2, b2y2) - fminf(b1y1, b2y1);
    const float c2 = cw * cw + ch * ch + EPSF;

    // ((b2x1+b2x2-b1x1-b1x2)^2 + (..y..)^2)/4  ==  (x2-x1)^2 + (y2-y1)^2
    const float dx = x2 - x1, dy = y2 - y1;
    const float rho2 = dx * dx + dy * dy;

    const float da = atanf(w2 / (h2 + EPSF)) - atanf(w1 / (h1 + EPSF));
    const float v  = FOUR_OVER_PI2 * da * da;
    const float alpha = v / (v - iou + (1.0f + EPSF));

    const float ciou = iou - (rho2 / c2 + v * alpha);
    return 1.0f - ciou;
}

__global__ void __launch_bounds__(256)
ciou_partial_kernel(const float4* __restrict__ preds,
                    const float4* __restrict__ tgts,
                    float* __restrict__ partials, int n) {
    const int tid    = (int)(blockIdx.x * blockDim.x + threadIdx.x);
    const int stride = (int)(gridDim.x * blockDim.x);

    // Uniform trip count: scalar loop condition, no per-iteration EXEC churn.
    const int full = n / stride;          // same for every thread
    const int rem  = n - full * stride;   // tail elements

    float acc = 0.0f;
    int i = tid;
    #pragma unroll 2
    for (int k = 0; k < full; ++k, i += stride) {
        acc += ciou_loss_elem(preds[i], tgts[i]);
    }
    if (tid < rem) {
        const int j = full * stride + tid;
        acc += ciou_loss_elem(preds[j], tgts[j]);
    }

    // EXEC fully re-converged here -> WMMA legal.
    const float wsum = wave_sum_wmma(acc);

    __shared__ float smem[8];
    const int wid  = threadIdx.x >> 5;
    const int lane = threadIdx.x & 31;
    if (lane == 0) smem[wid] = wsum;
    __syncthreads();
    if (threadIdx.x == 0) {
        float s = 0.0f;
        const int nw = (int)(blockDim.x >> 5);
        for (int q = 0; q < nw; ++q) s += smem[q];
        partials[blockIdx.x] = s;
    }
}

__global__ void __launch_bounds__(256)
ciou_final_kernel(const float* __restrict__ partials, int nblocks,
                  float* __restrict__ out, float inv_m) {
    float acc = 0.0f;
    for (int i = (int)threadIdx.x; i < nblocks; i += (int)blockDim.x)
        acc += partials[i];

    const float wsum = wave_sum_wmma(acc);

    __shared__ float smem[8];
    const int wid  = threadIdx.x >> 5;
    const int lane = threadIdx.x & 31;
    if (lane == 0) smem[wid] = wsum;
    __syncthreads();
    if (threadIdx.x == 0) {
        float s = 0.0f;
        for (int q = 0; q < 8; ++q) s += smem[q];
        out[0] = s * inv_m;
    }
}

extern "C" void kernel_launch(void* const* d_in, const int* in_sizes, int n_in,
                              void* d_out, int out_size, void* d_ws, size_t ws_size,
                              hipStream_t stream) {
    const float4* preds = (const float4*)d_in[0];
    const float4* tgts  = (const float4*)d_in[1];
    float* out      = (float*)d_out;
    float* partials = (float*)d_ws;

    const int n = in_sizes[0] / 4;            // number of boxes (M)
    int blocks = (n + 255) / 256;
    if (blocks > 2048) blocks = 2048;
    if (blocks < 1) blocks = 1;

    ciou_partial_kernel<<<blocks, 256, 0, stream>>>(preds, tgts, partials, n);
    ciou_final_kernel<<<1, 256, 0, stream>>>(partials, blocks, out,
                                             n > 0 ? 1.0f / (float)n : 0.0f);
}